// GNN_node_cross_62225486184591
// MI455X (gfx1250) — compile-verified
//
#include <hip/hip_runtime.h>
#include <hip/hip_bf16.h>

#define NN 20000
#define EE 320000
#define DD 128
#define LL 5
#define EV 5
#define ND (NN * DD)

typedef __attribute__((ext_vector_type(2))) float v2f;
typedef __attribute__((ext_vector_type(8))) float v8f;

// ---------------------------------------------------------------- embedding
__global__ __launch_bounds__(256) void embed_kernel(
    const int* __restrict__ x, const float* __restrict__ ne1,
    const float* __restrict__ ne2, float* __restrict__ h1,
    float* __restrict__ h2) {
  int idx = blockIdx.x * 256 + threadIdx.x;  // [0, N*32) exact
  int i = idx >> 5, c = idx & 31;
  int v = x[i];
  ((float4*)h1)[i * 32 + c] = ((const float4*)ne1)[v * 32 + c];
  ((float4*)h2)[i * 32 + c] = ((const float4*)ne2)[v * 32 + c];
}

// ---------------------------------------------------------------- zero
__global__ __launch_bounds__(256) void zero_kernel(float* __restrict__ p, int n) {
  int i = blockIdx.x * 256 + threadIdx.x;
  if (i < n) p[i] = 0.0f;
}

// ------------------------------------------------- edge message + scatter-add
// msg = relu(h[src] + edge_tab[eattr]); agg[dst] += msg  (native f32 atomics)
__global__ __launch_bounds__(256) void edge_kernel(
    const int* __restrict__ src, const int* __restrict__ dst,
    const int* __restrict__ eat, const float* __restrict__ h,
    const float* __restrict__ etab, float* __restrict__ agg) {
  int idx = blockIdx.x * 256 + threadIdx.x;  // [0, E*32) exact
  int e = idx >> 5, c = idx & 31;
  int s = src[e], d = dst[e], a = eat[e];
  float4 hv = ((const float4*)h)[s * 32 + c];
  float4 ev = ((const float4*)etab)[a * 32 + c];
  float m0 = fmaxf(hv.x + ev.x, 0.0f);
  float m1 = fmaxf(hv.y + ev.y, 0.0f);
  float m2 = fmaxf(hv.z + ev.z, 0.0f);
  float m3 = fmaxf(hv.w + ev.w, 0.0f);
  float* ap = agg + (size_t)d * DD + c * 4;
  unsafeAtomicAdd(ap + 0, m0);
  unsafeAtomicAdd(ap + 1, m1);
  unsafeAtomicAdd(ap + 2, m2);
  unsafeAtomicAdd(ap + 3, m3);
}

// ------------------------------------------------- GEMM (fp32 WMMA) + BN stats
// z = ((1+eps)*h + agg) @ W^T + b ; accumulates per-feature sum / sumsq.
// One wave computes a 16x16 tile via 32x V_WMMA_F32_16X16X4_F32.
__global__ __launch_bounds__(256) void gemm_stats_kernel(
    const float* __restrict__ h, const float* __restrict__ agg,
    const float* __restrict__ Wm, const float* __restrict__ bvec,
    const float* __restrict__ epsP, float* __restrict__ z,
    float* __restrict__ sumOut, float* __restrict__ sqOut) {
  const int lane = threadIdx.x & 31;
  const int wave = threadIdx.x >> 5;
  const int row0 = blockIdx.x * 128 + wave * 16;  // 16-row tile per wave
  const float ep = 1.0f + epsP[0];
  const int rA = row0 + (lane & 15);       // A-fragment source row
  const int kh = (lane >> 4);              // 0: k-pair {0,1}, 1: k-pair {2,3}
  const int nl = lane & 15;

  // Build A fragments (X = (1+eps)*h + agg), kept in registers for all 8 tiles
  v2f xa[32];
  if (rA < NN) {
    const v2f* hp = (const v2f*)(h + (size_t)rA * DD);
    const v2f* ap = (const v2f*)(agg + (size_t)rA * DD);
#pragma unroll
    for (int kk = 0; kk < 32; ++kk) {
      int c2 = kk * 2 + kh;  // (kk*4 + 2*kh)/2
      v2f hv = hp[c2], av = ap[c2];
      v2f r;
      r.x = ep * hv.x + av.x;
      r.y = ep * hv.y + av.y;
      xa[kk] = r;
    }
  } else {
#pragma unroll
    for (int kk = 0; kk < 32; ++kk) {
      v2f zz; zz.x = 0.0f; zz.y = 0.0f;
      xa[kk] = zz;
    }
  }

  for (int nt = 0; nt < 8; ++nt) {
    const int n0 = nt * 16;
    // B[k][j] = W[j][k]: lane reads contiguous k-pairs from W row (n0+nl)
    const v2f* wp = (const v2f*)(Wm + (size_t)(n0 + nl) * DD);
    v8f acc = {0.f, 0.f, 0.f, 0.f, 0.f, 0.f, 0.f, 0.f};
#if __has_builtin(__builtin_amdgcn_wmma_f32_16x16x4_f32)
#pragma unroll
    for (int kk = 0; kk < 32; ++kk) {
      v2f bfrag = wp[kk * 2 + kh];
      acc = __builtin_amdgcn_wmma_f32_16x16x4_f32(
          false, xa[kk], false, bfrag, (short)0, acc, false, false);
    }
#else
    // Fallback (should not trigger on gfx1250): scalar accumulate in C layout
    for (int v = 0; v < 8; ++v) {
      int m = row0 + v + (kh << 3);
      float s = 0.0f;
      if (m < NN)
        for (int k = 0; k < DD; ++k)
          s += (ep * h[(size_t)m * DD + k] + agg[(size_t)m * DD + k]) *
               Wm[(size_t)(n0 + nl) * DD + k];
      acc[v] = s;
    }
#endif
    // Epilogue: bias, store, per-column BN statistics (one column per lane)
    const float bias = bvec[n0 + nl];
    float lsum = 0.0f, lsq = 0.0f;
#pragma unroll
    for (int v = 0; v < 8; ++v) {
      int m = row0 + v + (kh << 3);  // C layout: lanes>=16 hold rows +8
      float val = acc[v] + bias;
      if (m < NN) {
        z[(size_t)m * DD + n0 + nl] = val;
        lsum += val;
        lsq += val * val;
      }
    }
    unsafeAtomicAdd(&sumOut[n0 + nl], lsum);
    unsafeAtomicAdd(&sqOut[n0 + nl], lsq);
  }
}

// ---------------------------------------------------------------- BN finalize
// stats: [sum1|sq1|sum2|sq2] (4x128) -> bnp: [scale1|shift1|scale2|shift2]
__global__ __launch_bounds__(256) void bn_finalize_kernel(
    const float* __restrict__ stats, const float* __restrict__ g1,
    const float* __restrict__ be1, const float* __restrict__ g2,
    const float* __restrict__ be2, float* __restrict__ bnp) {
  int t = threadIdx.x >> 7, j = threadIdx.x & 127;
  float s = stats[t * 256 + j];
  float q = stats[t * 256 + 128 + j];
  const float inv = 1.0f / (float)NN;
  float mu = s * inv;
  float var = q * inv - mu * mu;  // biased variance, matches jnp.var
  float g = t ? g2[j] : g1[j];
  float be = t ? be2[j] : be1[j];
  float sc = g * rsqrtf(var + 1e-5f);
  bnp[t * 256 + j] = sc;
  bnp[t * 256 + 128 + j] = be - mu * sc;
}

// --------------------------------------------- BN apply + relu + cross-stitch
__global__ __launch_bounds__(256) void bn_cross_kernel(
    const float* __restrict__ z1, const float* __restrict__ z2,
    const float* __restrict__ bnp, const float* __restrict__ cp, int doRelu,
    float* __restrict__ o1, float* __restrict__ o2) {
  int i4 = blockIdx.x * 256 + threadIdx.x;  // [0, ND/4) exact
  int jf = (i4 & 31) * 4;
  float4 a = ((const float4*)z1)[i4];
  float4 b = ((const float4*)z2)[i4];
  float c00 = cp[0], c01 = cp[1], c10 = cp[2], c11 = cp[3];
  float r1[4] = {a.x, a.y, a.z, a.w};
  float r2[4] = {b.x, b.y, b.z, b.w};
  float w1[4], w2[4];
#pragma unroll
  for (int u = 0; u < 4; ++u) {
    int j = jf + u;
    float a1 = r1[u] * bnp[j] + bnp[128 + j];
    float a2 = r2[u] * bnp[256 + j] + bnp[384 + j];
    if (doRelu) {
      a1 = fmaxf(a1, 0.0f);
      a2 = fmaxf(a2, 0.0f);
    }
    float n1 = c00 * a1 + c01 * a2;
    float n2 = c10 * n1 + c11 * a2;  // uses updated n1 (faithful to source)
    w1[u] = n1;
    w2[u] = n2;
  }
  float4 v1, v2;
  v1.x = w1[0]; v1.y = w1[1]; v1.z = w1[2]; v1.w = w1[3];
  v2.x = w2[0]; v2.y = w2[1]; v2.z = w2[2]; v2.w = w2[3];
  ((float4*)o1)[i4] = v1;
  ((float4*)o2)[i4] = v2;
}

// ---------------------------------------------------------------- launch
extern "C" void kernel_launch(void* const* d_in, const int* in_sizes, int n_in,
                              void* d_out, int out_size, void* d_ws,
                              size_t ws_size, hipStream_t stream) {
  const int* x = (const int*)d_in[0];
  const int* src = (const int*)d_in[1];       // edge_index[0]
  const int* dst = src + EE;                  // edge_index[1]
  const int* eat = (const int*)d_in[2];
  const float* ne1 = (const float*)d_in[3];
  const float* ne2 = (const float*)d_in[4];
  const float* eemb = (const float*)d_in[5];  // (2,L,EV,D)
  const float* epsA = (const float*)d_in[6];  // (2,L)
  const float* Wm = (const float*)d_in[7];    // (2,L,D,D)
  const float* bb = (const float*)d_in[8];    // (2,L,D)
  const float* gam = (const float*)d_in[9];   // (2,L,D)
  const float* bet = (const float*)d_in[10];  // (2,L,D)
  const float* cru = (const float*)d_in[11];  // (L,2,2)

  float* ws = (float*)d_ws;
  float* h1 = ws;
  float* h2 = ws + (size_t)ND;
  float* z1 = ws + 2 * (size_t)ND;
  float* z2 = ws + 3 * (size_t)ND;
  float* agg1 = ws + 4 * (size_t)ND;
  float* agg2 = ws + 5 * (size_t)ND;   // contiguous with agg1
  float* stats = ws + 6 * (size_t)ND;  // 512 floats, contiguous with agg2
  float* bnp = stats + 512;            // 512 floats

  embed_kernel<<<(NN * 32) / 256, 256, 0, stream>>>(x, ne1, ne2, h1, h2);

  for (int l = 0; l < LL; ++l) {
    const int zn = 2 * ND + 512;  // agg1 + agg2 + stats
    zero_kernel<<<(zn + 255) / 256, 256, 0, stream>>>(agg1, zn);

    edge_kernel<<<(EE * 32) / 256, 256, 0, stream>>>(
        src, dst, eat, h1, eemb + (size_t)(0 * LL + l) * EV * DD, agg1);
    edge_kernel<<<(EE * 32) / 256, 256, 0, stream>>>(
        src, dst, eat, h2, eemb + (size_t)(1 * LL + l) * EV * DD, agg2);

    gemm_stats_kernel<<<(NN + 127) / 128, 256, 0, stream>>>(
        h1, agg1, Wm + (size_t)(0 * LL + l) * DD * DD, bb + (0 * LL + l) * DD,
        epsA + (0 * LL + l), z1, stats + 0, stats + 128);
    gemm_stats_kernel<<<(NN + 127) / 128, 256, 0, stream>>>(
        h2, agg2, Wm + (size_t)(1 * LL + l) * DD * DD, bb + (1 * LL + l) * DD,
        epsA + (1 * LL + l), z2, stats + 256, stats + 384);

    bn_finalize_kernel<<<1, 256, 0, stream>>>(
        stats, gam + (0 * LL + l) * DD, bet + (0 * LL + l) * DD,
        gam + (1 * LL + l) * DD, bet + (1 * LL + l) * DD, bnp);

    float* o1 = (l == LL - 1) ? (float*)d_out : h1;
    float* o2 = (l == LL - 1) ? ((float*)d_out + ND) : h2;
    bn_cross_kernel<<<(ND / 4) / 256, 256, 0, stream>>>(
        z1, z2, bnp, cru + l * 4, (l < LL - 1) ? 1 : 0, o1, o2);
  }
}